// ViscoelasticMaterialModel_81432579932689
// MI455X (gfx1250) — compile-verified
//
#include <hip/hip_runtime.h>
#include <math.h>

#define FEAT 501
#define NIV 8
#define HID 64
#define DT 0.01f
#define KPAD 504          // per-half padded K (2*504 = 1008 >= 1002, zero-padded)
#define SM_STRIDE 508     // LDS row stride (bank-friendly: 508 % 64 = 60)
#define TC 16             // time-chunk for recurrence staging
#define NB 64             // batch rows per block in recurrence kernel

typedef __attribute__((ext_vector_type(2))) float v2f;
typedef __attribute__((ext_vector_type(8))) float v8f;

__device__ __forceinline__ float softplus_f(float x) {
    return (x > 20.0f) ? x : log1pf(expf(x));
}

// ---------------------------------------------------------------------------
// Kernel 1: per-batch time-invariant coefficients.
//   Bm   = relu(m@Bw1+Bb1)@Bw2+Bb2            [B,8]
//   beta = softplus(softplus(m@bw1+bb1)@bw2+bb2)  [B,8]
//   Bsum, E_p, nu_p                           [B]
// GEMM1 (K=1002) done with V_WMMA_F32_16X16X4_F32, 16-row batch tile / block,
// 4 waves = 4 column tiles of the 64 hidden units, both nets share the A tile.
// ---------------------------------------------------------------------------
__global__ __launch_bounds__(128)
void coeff_kernel(const float* __restrict__ E, const float* __restrict__ nu,
                  const float* __restrict__ Bw1, const float* __restrict__ Bb1,
                  const float* __restrict__ Bw2, const float* __restrict__ Bb2,
                  const float* __restrict__ bw1, const float* __restrict__ bb1,
                  const float* __restrict__ bw2, const float* __restrict__ bb2,
                  float* __restrict__ wsBm, float* __restrict__ wsBeta,
                  float* __restrict__ wsBsum, float* __restrict__ wsEp,
                  float* __restrict__ wsNup)
{
    __shared__ float sm[16 * SM_STRIDE];   // staged m-half tile [16][504]
    __shared__ float hB[16 * 68];          // relu hidden
    __shared__ float hb[16 * 68];          // softplus hidden
    __shared__ float redInv[16 * 9];
    __shared__ float redE[16 * 9];
    __shared__ float sBm[16 * 9];

    const int tid = threadIdx.x;
    const int b0 = blockIdx.x * 16;

    // ---- nu_p / E_p reductions (8 partial lanes per batch row) ----
    {
        int r = tid >> 3, g = tid & 7;
        const float* nuRow = nu + (size_t)(b0 + r) * FEAT;
        const float* ERow  = E  + (size_t)(b0 + r) * FEAT;
        float sInv = 0.f, sE = 0.f;
        for (int j = g; j < FEAT; j += 8) {
            float nv = nuRow[j];
            float inv = 1.0f / nv;
            sInv += inv;
            sE   += ERow[j] * inv * inv;
        }
        redInv[r * 9 + g] = sInv;
        redE[r * 9 + g]   = sE;
    }
    __syncthreads();
    if (tid < 16) {
        float sInv = 0.f, sE = 0.f;
        for (int g = 0; g < 8; ++g) { sInv += redInv[tid*9+g]; sE += redE[tid*9+g]; }
        float nup = (float)FEAT / sInv;                 // 1/mean(1/nu)
        float Ep  = (sE / (float)FEAT) * nup * nup;     // mean(E/nu^2)*nup^2
        wsNup[b0 + tid] = nup;
        wsEp[b0 + tid]  = Ep;
    }

    // ---- GEMM1: K swept in two zero-padded halves of 504 ----
    const int lane = tid & 31;
    const int wave = tid >> 5;
    const int n0   = wave * 16;          // hidden-column tile
    const int rowA = lane & 15;
    const int koff = (lane >> 4) << 1;   // lanes 0-15 -> K 0,1 ; lanes 16-31 -> K 2,3
    const int colB = lane & 15;

    v8f accB = {};   // relu net accumulator (16x16 tile)
    v8f accb = {};   // softplus net accumulator

    for (int half = 0; half < 2; ++half) {
        __syncthreads();
        // stage m = concat(E, nu) half-tile into LDS (coalesced)
        for (int idx = tid; idx < 16 * KPAD; idx += 128) {
            int row = idx / KPAD;
            int jj  = idx - row * KPAD;
            int j   = half * KPAD + jj;
            float v = 0.f;
            if (j < FEAT)          v = E[(size_t)(b0+row)*FEAT + j];
            else if (j < 2*FEAT)   v = nu[(size_t)(b0+row)*FEAT + (j - FEAT)];
            sm[row * SM_STRIDE + jj] = v;
        }
        __syncthreads();

        for (int jj = 0; jj < KPAD; jj += 4) {
            int j0 = half * KPAD + jj + koff;
            v2f a;
            a.x = sm[rowA * SM_STRIDE + jj + koff];
            a.y = sm[rowA * SM_STRIDE + jj + koff + 1];
            v2f bB, bbv;
            bB.x  = (j0     < 2*FEAT) ? Bw1[(size_t)j0      * HID + n0 + colB] : 0.f;
            bB.y  = (j0 + 1 < 2*FEAT) ? Bw1[(size_t)(j0+1)  * HID + n0 + colB] : 0.f;
            bbv.x = (j0     < 2*FEAT) ? bw1[(size_t)j0      * HID + n0 + colB] : 0.f;
            bbv.y = (j0 + 1 < 2*FEAT) ? bw1[(size_t)(j0+1)  * HID + n0 + colB] : 0.f;
            accB = __builtin_amdgcn_wmma_f32_16x16x4_f32(
                       false, a, false, bB,  (short)0, accB, false, false);
            accb = __builtin_amdgcn_wmma_f32_16x16x4_f32(
                       false, a, false, bbv, (short)0, accb, false, false);
        }
    }

    // ---- bias + activation -> hidden tiles in LDS ----
    {
        int mBase = (lane >> 4) << 3;        // C/D layout: VGPR r -> rows r / r+8
        int col = n0 + colB;
        float biasB = Bb1[col];
        float biasb = bb1[col];
#pragma unroll
        for (int r = 0; r < 8; ++r) {
            int M = r + mBase;
            float xB = accB[r] + biasB;
            hB[M * 68 + col] = xB > 0.f ? xB : 0.f;
            hb[M * 68 + col] = softplus_f(accb[r] + biasb);
        }
    }
    __syncthreads();

    // ---- GEMM2 (64->8), thread (row,k), then outputs ----
    {
        int row = tid >> 3, k = tid & 7;
        float aB = Bb2[k], ab = bb2[k];
        for (int c = 0; c < HID; ++c) {
            aB += hB[row * 68 + c] * Bw2[c * NIV + k];
            ab += hb[row * 68 + c] * bw2[c * NIV + k];
        }
        wsBm[(size_t)(b0 + row) * NIV + k]   = aB;
        wsBeta[(size_t)(b0 + row) * NIV + k] = softplus_f(ab);
        sBm[row * 9 + k] = aB;
    }
    __syncthreads();
    if (tid < 16) {
        float s = 0.f;
        for (int k = 0; k < NIV; ++k) s += sBm[tid * 9 + k];
        wsBsum[b0 + tid] = s;
    }
}

// ---------------------------------------------------------------------------
// Kernel 2: serial time recurrence, one thread per batch row, xi[8] in regs.
// e/e_dot chunks staged into LDS with coalesced float4 loads; stress and the
// dominant xi output (134 MB) staged in LDS and written as contiguous float4
// rows (full HBM line utilization). Padded LDS strides avoid bank conflicts.
// ---------------------------------------------------------------------------
__global__ __launch_bounds__(64)
void recur_kernel(const float* __restrict__ e, const float* __restrict__ edot,
                  const float* __restrict__ wsBm, const float* __restrict__ wsBeta,
                  const float* __restrict__ wsBsum, const float* __restrict__ wsEp,
                  const float* __restrict__ wsNup,
                  float* __restrict__ outStress, float* __restrict__ outXi,
                  int T)
{
    __shared__ float se[NB * 17];
    __shared__ float spd[NB * 17];
    __shared__ float ss[NB * 17];
    __shared__ float sxi[NB * 129];     // [64 rows][16*8 + 1 pad]

    const int tid = threadIdx.x;
    const int b0 = blockIdx.x * NB;
    const int b  = b0 + tid;

    float Bm[NIV], beta[NIV], xi[NIV];
#pragma unroll
    for (int k = 0; k < NIV; ++k) {
        Bm[k]   = wsBm[(size_t)b * NIV + k];
        beta[k] = wsBeta[(size_t)b * NIV + k];
        xi[k]   = 0.f;
    }
    const float Bsum = wsBsum[b];
    const float Ep   = wsEp[b];
    const float nup  = wsNup[b];
    const float ce   = Bsum * Ep;

    for (int t0 = 0; t0 < T; t0 += TC) {
        // stage e / e_dot chunk (coalesced float4 reads)
        for (int v = tid; v < NB * (TC/4); v += NB) {
            int row = v >> 2, seg = v & 3;
            float4 ev = *(const float4*)(e    + (size_t)(b0+row)*T + t0 + seg*4);
            float4 pv = *(const float4*)(edot + (size_t)(b0+row)*T + t0 + seg*4);
            int base = row * 17 + seg * 4;
            se[base+0]=ev.x; se[base+1]=ev.y; se[base+2]=ev.z; se[base+3]=ev.w;
            spd[base+0]=pv.x; spd[base+1]=pv.y; spd[base+2]=pv.z; spd[base+3]=pv.w;
        }
        __syncthreads();

        // 16 serial recurrence steps for this thread's batch row
        for (int tt = 0; tt < TC; ++tt) {
            float u = se[tid*17 + tt];
            float p = spd[tid*17 + tt];
            float s = 0.f;
#pragma unroll
            for (int k = 0; k < NIV; ++k) {
                sxi[tid*129 + tt*8 + k] = xi[k];       // emit pre-update xi
                float du = u - xi[k];
                s += Bm[k] * du;
                float q = Bm[k] * du - Bsum * xi[k];   // q = -grad_v
                xi[k] += DT * beta[k] * q;             // xi += DT*beta*q
            }
            // stress = Bsum*E_p*u + sum(Bm*du) + nu_p*p   (= grad_u - grad_p)
            ss[tid*17 + tt] = ce * u + s + nup * p;
        }
        __syncthreads();

        // write stress chunk (contiguous float4 per row)
        for (int v = tid; v < NB * (TC/4); v += NB) {
            int row = v >> 2, seg = v & 3;
            int base = row * 17 + seg * 4;
            float4 sv = { ss[base], ss[base+1], ss[base+2], ss[base+3] };
            *(float4*)(outStress + (size_t)(b0+row)*T + t0 + seg*4) = sv;
        }
        // write xi chunk: 64 rows x 128 contiguous floats each
        for (int v = tid; v < NB * 32; v += NB) {
            int row = v >> 5, off = (v & 31) * 4;
            float4 xv = { sxi[row*129 + off],     sxi[row*129 + off + 1],
                          sxi[row*129 + off + 2], sxi[row*129 + off + 3] };
            *(float4*)(outXi + (size_t)(b0+row)*T*NIV + (size_t)t0*NIV + off) = xv;
        }
        __syncthreads();
    }
}

extern "C" void kernel_launch(void* const* d_in, const int* in_sizes, int n_in,
                              void* d_out, int out_size, void* d_ws, size_t ws_size,
                              hipStream_t stream)
{
    const float* e    = (const float*)d_in[0];
    const float* edot = (const float*)d_in[1];
    const float* E    = (const float*)d_in[2];
    const float* nu   = (const float*)d_in[3];
    const float* Bw1  = (const float*)d_in[4];
    const float* Bb1  = (const float*)d_in[5];
    const float* Bw2  = (const float*)d_in[6];
    const float* Bb2  = (const float*)d_in[7];
    const float* bw1  = (const float*)d_in[8];
    const float* bb1  = (const float*)d_in[9];
    const float* bw2  = (const float*)d_in[10];
    const float* bb2  = (const float*)d_in[11];

    const int B = in_sizes[2] / FEAT;    // 8192
    const int T = in_sizes[0] / B;       // 512

    float* ws = (float*)d_ws;
    float* wsBm   = ws;
    float* wsBeta = ws + (size_t)B * NIV;
    float* wsBsum = ws + (size_t)2 * B * NIV;
    float* wsEp   = wsBsum + B;
    float* wsNup  = wsEp + B;

    float* outStress = (float*)d_out;
    float* outXi     = outStress + (size_t)B * T;

    coeff_kernel<<<B / 16, 128, 0, stream>>>(E, nu, Bw1, Bb1, Bw2, Bb2,
                                             bw1, bb1, bw2, bb2,
                                             wsBm, wsBeta, wsBsum, wsEp, wsNup);
    recur_kernel<<<B / NB, NB, 0, stream>>>(e, edot, wsBm, wsBeta, wsBsum,
                                            wsEp, wsNup, outStress, outXi, T);
}